// RNN_84061099917346
// MI455X (gfx1250) — compile-verified
//
#include <hip/hip_runtime.h>
#include <math.h>

// MI455X (gfx1250) implementation of the Mogrifier-RNN decoder.
// fp32 WMMA (V_WMMA_F32_16X16X4_F32) everywhere: batch B=16 == WMMA M=16,
// so every skinny recurrent GEMM maps to one 16x16 accumulator tile per wave.
// Weights (~112MB) and E (128MB) are L2-resident (192MB L2), so the serial
// recurrence is L2-BW bound and fp32 (exact vs reference) costs little.

typedef float v2f __attribute__((ext_vector_type(2)));
typedef float v8f __attribute__((ext_vector_type(8)));

#define Hd   1024
#define Bd   16
#define Td   256
#define Ld   2
#define Rd   3
#define Vd   32000
#define EPSf 1e-6f

// ---------------------------------------------------------------- embedding
__global__ __launch_bounds__(256) void rnn_embed(const float* __restrict__ E,
                                                 const int* __restrict__ x,
                                                 float* __restrict__ xs) {
  // grid = T*B blocks; xs is time-major [T, B, H]
  int blk = blockIdx.x;
  int t = blk / Bd, b = blk % Bd;
  int idx = x[b * Td + t];
  const float4* src = (const float4*)(E + (size_t)idx * Hd);
  float4* dst = (float4*)(xs + (size_t)blk * Hd);
  dst[threadIdx.x] = src[threadIdx.x];   // 256 threads * 4 floats = 1024
}

// ---------------------------------------------------------------- zero init
__global__ __launch_bounds__(256) void rnn_zero(float* p, int n) {
  int i = blockIdx.x * 256 + threadIdx.x;
  if (i < n) p[i] = 0.f;
}

// ---------------------------------------------------------------- layernorm
__global__ __launch_bounds__(256) void rnn_ln(const float* __restrict__ x,
                                              const float* __restrict__ gamma,
                                              const float* __restrict__ beta,
                                              float* __restrict__ out) {
  __shared__ float red[256];
  int b = blockIdx.x;
  const float* xr = x + (size_t)b * Hd;
  float v[4];
  float s = 0.f;
#pragma unroll
  for (int i = 0; i < 4; ++i) { v[i] = xr[threadIdx.x + i * 256]; s += v[i]; }
  red[threadIdx.x] = s; __syncthreads();
  for (int off = 128; off > 0; off >>= 1) {
    if (threadIdx.x < off) red[threadIdx.x] += red[threadIdx.x + off];
    __syncthreads();
  }
  float mean = red[0] * (1.f / Hd);
  __syncthreads();
  s = 0.f;
#pragma unroll
  for (int i = 0; i < 4; ++i) { float d = v[i] - mean; s += d * d; }
  red[threadIdx.x] = s; __syncthreads();
  for (int off = 128; off > 0; off >>= 1) {
    if (threadIdx.x < off) red[threadIdx.x] += red[threadIdx.x + off];
    __syncthreads();
  }
  float sigma = sqrtf(red[0] * (1.f / Hd));
  float inv = 1.f / (sigma + EPSf);
#pragma unroll
  for (int i = 0; i < 4; ++i) {
    int col = threadIdx.x + i * 256;
    out[(size_t)b * Hd + col] = gamma[col] * ((v[i] - mean) * inv) + beta[col];
  }
}

// ------------------------------------------------- skinny GEMM (M = 16) + epilogue
// C[16,N] = act( A[16,K] * W[K,N] + bias0 + bias1 ) (+adds) or gate*2*sigmoid(z)
// A is a concat of A0 (k < Ksplit) and A1 (k >= Ksplit), both leading-dim ldA;
// pass Ksplit == K for a plain single-operand GEMM.
// One wave per 16x16 output tile via V_WMMA_F32_16X16X4_F32. 4 waves/block.
// Inner loop: b64 A-frag load + 2 b32 W loads + prefetch 16 rows ahead + 1 WMMA.
__global__ __launch_bounds__(128) void rnn_gemm16(
    const float* __restrict__ A0, const float* __restrict__ A1, int Ksplit,
    int K, int ldA,
    const float* __restrict__ W, int N,
    const float* __restrict__ bias0, const float* __restrict__ bias1,
    const float* gate,        // non-null => out = gate * 2*sigmoid(z)
    int act,                  // 0 none, 1 tanh, 2 relu
    const float* __restrict__ addA, const float* __restrict__ addB,
    const float* __restrict__ addC,
    float* out) {
  const int wave = threadIdx.x >> 5;
  const int lane = threadIdx.x & 31;
  const int row  = lane & 15;
  const int off  = (lane >> 4) << 1;           // 0 (lanes 0-15) or 2 (16-31)
  const int n0   = (blockIdx.x * 4 + wave) * 16;
  const int n    = n0 + row;

  v8f c = {0.f, 0.f, 0.f, 0.f, 0.f, 0.f, 0.f, 0.f};

  const int K1 = (Ksplit < K) ? Ksplit : K;
  const size_t bstep = (size_t)4 * N;          // 4 W rows per WMMA step
  const size_t pfoff = (size_t)16 * N;         // prefetch 16 rows (4 iters) ahead
  const float* ap = A0 + (size_t)row * ldA + off;
  const float* bp = W + (size_t)off * N + n;

#pragma unroll 4
  for (int k = 0; k < K1; k += 4) {
    v2f a, bb;
    a.x = ap[0];                 // A[row][k+off]     -> VGPR0
    a.y = ap[1];                 // A[row][k+off+1]   -> VGPR1
    bb.x = bp[0];                // W[k+off][n]
    bb.y = bp[N];                // W[k+off+1][n]
    __builtin_prefetch(bp + pfoff, 0, 3);      // global_prefetch: next W rows
    c = __builtin_amdgcn_wmma_f32_16x16x4_f32(false, a, false, bb,
                                              (short)0, c, false, false);
    ap += 4;
    bp += bstep;
  }
  if (K1 < K) {                                // concat tail: switch A operand
    const float* ap2 = A1 + (size_t)row * ldA + off;
#pragma unroll 4
    for (int k = K1; k < K; k += 4) {
      v2f a, bb;
      a.x = ap2[0];
      a.y = ap2[1];
      bb.x = bp[0];
      bb.y = bp[N];
      __builtin_prefetch(bp + pfoff, 0, 3);
      c = __builtin_amdgcn_wmma_f32_16x16x4_f32(false, a, false, bb,
                                                (short)0, c, false, false);
      ap2 += 4;
      bp += bstep;
    }
  }

  const float bz = bias0[n] + (bias1 ? bias1[n] : 0.f);
  const int mbase = (lane >> 4) << 3;          // 0 or 8
#pragma unroll
  for (int r = 0; r < 8; ++r) {
    int m = mbase + r;
    size_t o = (size_t)m * N + n;
    float z = c[r] + bz;
    float res;
    if (gate) {
      res = gate[o] * 2.f * (1.f / (1.f + __expf(-z)));
    } else {
      if (act == 1)      res = tanhf(z);
      else if (act == 2) res = fmaxf(z, 0.f);
      else               res = z;
      if (addA) res += addA[o];
      if (addB) res += addB[o];
      if (addC) res += addC[o];
    }
    out[o] = res;
  }
}

// ------------------------------------------------- logits: [B*T,H] x E^T -> [B,T,V]
// 64x64 block tile, 8 waves (each 16x32 = two 16x16 WMMA tiles), LDS-staged,
// E transposed through LDS (NT product).
__global__ __launch_bounds__(256) void rnn_logits(
    const float* __restrict__ dec,   // [T*B, H] time-major rows
    const float* __restrict__ E,     // [V, H]
    float* __restrict__ out) {       // [B, T, V]
  __shared__ float As[64][17];       // +1 pad: conflict-free strided reads
  __shared__ float Bs[16][68];
  const int n0 = blockIdx.x * 64;
  const int m0 = blockIdx.y * 64;
  const int tid = threadIdx.x;
  const int wave = tid >> 5;
  const int lane = tid & 31;
  const int row = lane & 15;
  const int off = (lane >> 4) << 1;
  const int mt = (wave >> 1) * 16;
  const int nt = (wave & 1) * 32;

  v8f c0 = {0.f, 0.f, 0.f, 0.f, 0.f, 0.f, 0.f, 0.f};
  v8f c1 = {0.f, 0.f, 0.f, 0.f, 0.f, 0.f, 0.f, 0.f};

  const int lr = tid >> 2;            // 0..63
  const int lc = (tid & 3) * 4;       // 0,4,8,12

  for (int k0 = 0; k0 < Hd; k0 += 16) {
    float4 av = *(const float4*)(dec + (size_t)(m0 + lr) * Hd + k0 + lc);
    As[lr][lc + 0] = av.x; As[lr][lc + 1] = av.y;
    As[lr][lc + 2] = av.z; As[lr][lc + 3] = av.w;
    float4 ev = *(const float4*)(E + (size_t)(n0 + lr) * Hd + k0 + lc);
    Bs[lc + 0][lr] = ev.x; Bs[lc + 1][lr] = ev.y;
    Bs[lc + 2][lr] = ev.z; Bs[lc + 3][lr] = ev.w;
    __syncthreads();
#pragma unroll
    for (int kk = 0; kk < 16; kk += 4) {
      v2f a, b0, b1;
      a.x = As[mt + row][kk + off];
      a.y = As[mt + row][kk + off + 1];
      b0.x = Bs[kk + off][nt + row];
      b0.y = Bs[kk + off + 1][nt + row];
      b1.x = Bs[kk + off][nt + 16 + row];
      b1.y = Bs[kk + off + 1][nt + 16 + row];
      c0 = __builtin_amdgcn_wmma_f32_16x16x4_f32(false, a, false, b0,
                                                 (short)0, c0, false, false);
      c1 = __builtin_amdgcn_wmma_f32_16x16x4_f32(false, a, false, b1,
                                                 (short)0, c1, false, false);
    }
    __syncthreads();
  }

  const int mbase = (lane >> 4) << 3;
#pragma unroll
  for (int r = 0; r < 8; ++r) {
    int m = m0 + mt + mbase + r;      // global dec row = t*B + b
    int t = m >> 4;                   // /16
    int b = m & 15;
    size_t orow = ((size_t)b * Td + t) * (size_t)Vd;
    out[orow + n0 + nt + row] = c0[r];
    out[orow + n0 + nt + 16 + row] = c1[r];
  }
}

// ---------------------------------------------------------------- launcher
extern "C" void kernel_launch(void* const* d_in, const int* in_sizes, int n_in,
                              void* d_out, int out_size, void* d_ws, size_t ws_size,
                              hipStream_t stream) {
  (void)in_sizes; (void)n_in; (void)out_size; (void)ws_size;
  const float* E        = (const float*)d_in[0];
  const float* ln_gamma = (const float*)d_in[1];
  const float* ln_beta  = (const float*)d_in[2];
  const float* Wm_k     = (const float*)d_in[3];
  const float* Wm_b     = (const float*)d_in[4];
  const float* Wh_k     = (const float*)d_in[5];
  const float* Wh_b     = (const float*)d_in[6];
  const float* bh       = (const float*)d_in[7];
  const float* Wy_k     = (const float*)d_in[8];
  const float* Wy_b     = (const float*)d_in[9];
  const float* by       = (const float*)d_in[10];
  const float* W1_k     = (const float*)d_in[11];
  const float* W1_b     = (const float*)d_in[12];
  const float* W2_k     = (const float*)d_in[13];
  const float* W2_b     = (const float*)d_in[14];
  const int*   x        = (const int*)d_in[15];
  float* out = (float*)d_out;

  float* ws     = (float*)d_ws;
  float* x_emb  = ws;                                  // T*B*H
  float* dec    = x_emb + (size_t)Td * Bd * Hd;        // T*B*H
  float* h_st   = dec   + (size_t)Td * Bd * Hd;        // L*B*H
  float* x_norm = h_st  + (size_t)Ld * Bd * Hd;        // B*H
  float* x_mog  = x_norm + (size_t)Bd * Hd;            // B*H
  float* h_mog  = x_mog  + (size_t)Bd * Hd;            // B*H
  float* y_buf  = h_mog  + (size_t)Bd * Hd;            // B*H
  float* a1     = y_buf  + (size_t)Bd * Hd;            // B*4H
  float* out0   = a1     + (size_t)Bd * 4 * Hd;        // B*H

  rnn_zero<<<(Ld * Bd * Hd + 255) / 256, 256, 0, stream>>>(h_st, Ld * Bd * Hd);
  rnn_embed<<<Td * Bd, 256, 0, stream>>>(E, x, x_emb);

  for (int t = 0; t < Td; ++t) {
    const float* xs_t = x_emb + (size_t)t * Bd * Hd;
    float* dec_t = dec + (size_t)t * Bd * Hd;
    for (int m = 0; m < Ld; ++m) {
      const float* xin  = (m == 0) ? xs_t : out0;
      const float* prev = (m == 0) ? nullptr : xs_t;   // res_conn across layers
      float* outp = (m == 0) ? out0 : dec_t;
      float* hm   = h_st + (size_t)m * Bd * Hd;

      rnn_ln<<<Bd, 256, 0, stream>>>(xin, ln_gamma + m * Hd, ln_beta + m * Hd, x_norm);

      // mogrifier rounds: rd0 h_mog = h_prev*2s(x_norm@W0); rd1 x_mog = x_norm*2s(h_mog@W1);
      // rd2 h_mog = h_mog*2s(x_mog@W2)
      rnn_gemm16<<<Hd / 64, 128, 0, stream>>>(
          x_norm, x_norm, Hd, Hd, Hd, Wm_k + ((size_t)(m * Rd + 0)) * Hd * Hd, Hd,
          Wm_b + (m * Rd + 0) * Hd, nullptr, hm, 0, nullptr, nullptr, nullptr, h_mog);
      rnn_gemm16<<<Hd / 64, 128, 0, stream>>>(
          h_mog, h_mog, Hd, Hd, Hd, Wm_k + ((size_t)(m * Rd + 1)) * Hd * Hd, Hd,
          Wm_b + (m * Rd + 1) * Hd, nullptr, x_norm, 0, nullptr, nullptr, nullptr, x_mog);
      rnn_gemm16<<<Hd / 64, 128, 0, stream>>>(
          x_mog, x_mog, Hd, Hd, Hd, Wm_k + ((size_t)(m * Rd + 2)) * Hd * Hd, Hd,
          Wm_b + (m * Rd + 2) * Hd, nullptr, h_mog, 0, nullptr, nullptr, nullptr, h_mog);

      // h_curr = tanh([x_mog, h_mog] @ Wh + Wh_b + bh) -> overwrites h state
      rnn_gemm16<<<Hd / 64, 128, 0, stream>>>(
          x_mog, h_mog, Hd, 2 * Hd, Hd, Wh_k + (size_t)m * 2 * Hd * Hd, Hd,
          Wh_b + m * Hd, bh + m * Hd, nullptr, 1, nullptr, nullptr, nullptr, hm);

      // FFN up: a1 = relu(x_norm @ W1 + W1_b), N = 4H
      rnn_gemm16<<<(4 * Hd) / 64, 128, 0, stream>>>(
          x_norm, x_norm, Hd, Hd, Hd, W1_k + (size_t)m * Hd * 4 * Hd, 4 * Hd,
          W1_b + m * 4 * Hd, nullptr, nullptr, 2, nullptr, nullptr, nullptr, a1);

      // y = h_curr @ Wy + Wy_b + by
      rnn_gemm16<<<Hd / 64, 128, 0, stream>>>(
          hm, hm, Hd, Hd, Hd, Wy_k + (size_t)m * Hd * Hd, Hd,
          Wy_b + m * Hd, by + m * Hd, nullptr, 0, nullptr, nullptr, nullptr, y_buf);

      // out = a1 @ W2 + W2_b + y + x_curr (+ prev_input)
      rnn_gemm16<<<Hd / 64, 128, 0, stream>>>(
          a1, a1, 4 * Hd, 4 * Hd, 4 * Hd, W2_k + (size_t)m * 4 * Hd * Hd, Hd,
          W2_b + m * Hd, nullptr, nullptr, 0, y_buf, xin, prev, outp);
    }
  }

  dim3 g(Vd / 64, (Bd * Td) / 64);
  rnn_logits<<<g, 256, 0, stream>>>(dec, E, out);
}